// AdaptiveEmbedding_86517821214166
// MI455X (gfx1250) — compile-verified
//
#include <hip/hip_runtime.h>

typedef float v2f __attribute__((ext_vector_type(2)));
typedef float v8f __attribute__((ext_vector_type(8)));

#define DP 1024            // D_PROJ
#define NTOK_MAX 16384     // 4*4096 flat tokens

// ---------------- kernel 0: zero the bucket counters ----------------
__global__ void AE_zero_counts(unsigned* __restrict__ counts) {
    if (threadIdx.x < 4) counts[threadIdx.x] = 0u;
}

// ---------------- kernel 1: partition tokens into 4 bucket lists ----------------
// list entry packs: bits[13:0] = flat token position, bits[31:14] = local row idx in table
__global__ void AE_partition(const int* __restrict__ inp, int n,
                             unsigned* __restrict__ counts, int* __restrict__ lists) {
    int t = blockIdx.x * blockDim.x + threadIdx.x;
    if (t >= n) return;
    int v = inp[t];
    int b  = (v < 20000) ? 0 : (v < 40000) ? 1 : (v < 200000) ? 2 : 3;
    int lo = (b == 0) ? 0 : (b == 1) ? 20000 : (b == 2) ? 40000 : 200000;
    unsigned pos = atomicAdd(&counts[b], 1u);
    lists[b * n + pos] = t | ((v - lo) << 14);
}

// ---------------- kernel 2: per-bucket tiled GEMM via V_WMMA_F32_16X16X4_F32 --------
// grid: (DP/16 col tiles, n/16 row tiles, 4 buckets); block: 1 wave (32 threads).
// out[pos, p] = 32 * sum_k emb_b[idx, k] * proj_b[p, k]
__global__ void __launch_bounds__(32)
AE_wmma_gemm(const float* __restrict__ e0, const float* __restrict__ e1,
             const float* __restrict__ e2, const float* __restrict__ e3,
             const float* __restrict__ p0, const float* __restrict__ p1,
             const float* __restrict__ p2, const float* __restrict__ p3,
             const unsigned* __restrict__ counts, const int* __restrict__ lists,
             float* __restrict__ out, int n) {
    const int b = blockIdx.z;
    const unsigned cnt = counts[b];
    const int rbase = blockIdx.y * 16;
    if ((unsigned)rbase >= cnt) return;                 // block-uniform exit

    const int d = DP >> (2 * b);                        // 1024,256,64,16 (all % 8 == 0)
    const float* __restrict__ emb  = (b == 0) ? e0 : (b == 1) ? e1 : (b == 2) ? e2 : e3;
    const float* __restrict__ proj = (b == 0) ? p0 : (b == 1) ? p1 : (b == 2) ? p2 : p3;
    const int* __restrict__ tile   = lists + b * n + rbase;
    const int valid = min(16, (int)cnt - rbase);

    const int lane = threadIdx.x;                       // wave32
    const int half = lane >> 4;                         // 0: K={0,1}, 1: K={2,3}
    const int l16  = lane & 15;

    // A fragment source: emb row of token M = l16 (pad rows read row of entry 0)
    const int mA = (l16 < valid) ? l16 : 0;
    const unsigned packA = (unsigned)tile[mA];
    const float* __restrict__ arow = emb + (size_t)(packA >> 14) * (size_t)d + (half * 2);

    // B fragment source: proj row p = pbase + l16 (B = proj^T, row-major [DP, d])
    const int pbase = blockIdx.x * 16;
    const float* __restrict__ brow = proj + (size_t)(pbase + l16) * (size_t)d + (half * 2);

    v8f acc = {0.f, 0.f, 0.f, 0.f, 0.f, 0.f, 0.f, 0.f};

    // K-loop unrolled x2: issue all 4 loads, then 2 dependent WMMAs, so the
    // second pair of loads overlaps the first wmma's wait.
    for (int k0 = 0; k0 < d; k0 += 8) {
        v2f a0 = *(const v2f*)(arow + k0);
        v2f b0 = *(const v2f*)(brow + k0);
        v2f a1 = *(const v2f*)(arow + k0 + 4);
        v2f b1 = *(const v2f*)(brow + k0 + 4);
        // 8 args: (neg_a, A, neg_b, B, c_mod, C, reuse_a, reuse_b)
        acc = __builtin_amdgcn_wmma_f32_16x16x4_f32(
            false, a0, false, b0, (short)0, acc, false, false);
        acc = __builtin_amdgcn_wmma_f32_16x16x4_f32(
            false, a1, false, b1, (short)0, acc, false, false);
    }

    // D layout (16x16 f32): lanes 0-15 -> N=lane, M=r ; lanes 16-31 -> N=lane-16, M=8+r
    float* __restrict__ ocol = out + pbase + l16;
    if (valid == 16) {
        // common case: full tile, branch-free, EXEC stays all-ones
        #pragma unroll
        for (int r = 0; r < 8; ++r) {
            const unsigned pk = (unsigned)tile[r + 8 * half];
            ocol[(size_t)(pk & 0x3FFFu) * DP] = acc[r] * 32.0f;     // EMB_SCALE = 32
        }
    } else {
        // tail tile: mask padded rows
        #pragma unroll
        for (int r = 0; r < 8; ++r) {
            const int M = r + 8 * half;
            if (M < valid) {
                const unsigned pk = (unsigned)tile[M];
                ocol[(size_t)(pk & 0x3FFFu) * DP] = acc[r] * 32.0f;
            }
        }
    }
}

extern "C" void kernel_launch(void* const* d_in, const int* in_sizes, int n_in,
                              void* d_out, int out_size, void* d_ws, size_t ws_size,
                              hipStream_t stream) {
    const int*   inp = (const int*)d_in[0];
    const float* e0  = (const float*)d_in[1];
    const float* e1  = (const float*)d_in[2];
    const float* e2  = (const float*)d_in[3];
    const float* e3  = (const float*)d_in[4];
    const float* p0  = (const float*)d_in[5];
    const float* p1  = (const float*)d_in[6];
    const float* p2  = (const float*)d_in[7];
    const float* p3  = (const float*)d_in[8];
    float* out = (float*)d_out;

    const int n = in_sizes[0];                          // 16384 flat tokens

    unsigned* counts = (unsigned*)d_ws;                 // 4 counters
    int* lists = (int*)((char*)d_ws + 256);             // 4 * n packed entries

    AE_zero_counts<<<1, 32, 0, stream>>>(counts);
    AE_partition<<<(n + 255) / 256, 256, 0, stream>>>(inp, n, counts, lists);

    dim3 grid(DP / 16, (n + 15) / 16, 4);
    AE_wmma_gemm<<<grid, 32, 0, stream>>>(e0, e1, e2, e3, p0, p1, p2, p3,
                                          counts, lists, out, n);
}